// _ConvBlock1_43018392436821
// MI455X (gfx1250) — compile-verified
//
#include <hip/hip_runtime.h>

// ---------------------------------------------------------------------------
// ConvBlock graph network for MI455X (gfx1250, wave32).
// Scatter stages: float4 gathers + global_atomic_add_f32 (L2-resident).
// GEMM stages: fp32 WMMA (v_wmma_f32_16x16x4_f32), one 16x16 tile per wave.
// ---------------------------------------------------------------------------

typedef __attribute__((ext_vector_type(2))) float v2f;
typedef __attribute__((ext_vector_type(8))) float v8f;

#define N_CENTERS 50000
#define N_FACES   100000
#define N_POINTS  50000
#define E_CF      200000
#define E_FP      200000
#define E_PP      400000
#define C_IN      64
#define C_CF      128
#define C_CAT     192
#define C_OUT     256

// -------------------------------- zero fill --------------------------------
__global__ __launch_bounds__(256) void k_zero(float* __restrict__ p, int n4) {
    int t = blockIdx.x * blockDim.x + threadIdx.x;
    if (t < n4) ((float4*)p)[t] = make_float4(0.f, 0.f, 0.f, 0.f);
}

// --------------------------- edge scatter-sum ------------------------------
// msg = x[src] * ew ; agg[dst] += msg   (one thread per edge x 4-channel group)
__global__ __launch_bounds__(256) void k_scatter(
    const float* __restrict__ x, const int* __restrict__ src,
    const int* __restrict__ dst, const float* __restrict__ ew,
    float* __restrict__ agg, int E, int C4, int C) {
    int t = blockIdx.x * blockDim.x + threadIdx.x;
    if (t >= E * C4) return;
    int e = t / C4;
    int g = (t - e * C4) << 2;
    int s = src[e], d = dst[e];
    float w = ew[e];
    float4 v = *(const float4*)(x + (size_t)s * C + g);
    float* o = agg + (size_t)d * C + g;
    atomicAdd(o + 0, v.x * w);
    atomicAdd(o + 1, v.y * w);
    atomicAdd(o + 2, v.z * w);
    atomicAdd(o + 3, v.w * w);
}

// ------------------- copy xFace into xcat[:, 128:192] ----------------------
__global__ __launch_bounds__(256) void k_copy_face(const float* __restrict__ xf,
                                                   float* __restrict__ xcat) {
    int t = blockIdx.x * blockDim.x + threadIdx.x;   // N_FACES * 16 threads
    int f = t >> 4;
    int g = (t & 15) << 2;
    if (f >= N_FACES) return;
    float4 v = *(const float4*)(xf + (size_t)f * C_IN + g);
    *(float4*)(xcat + (size_t)f * C_CAT + C_CF + g) = v;
}

// --------------------------- fp32 WMMA GEMM --------------------------------
// C[row0:row0+16, col0:col0+16] = bias + A*B (+ A2*B2), row-major operands.
// One wave per 16x16 tile; 8 waves per block; all dims multiples of 16.
// A layout (ISA 7.12.2, 16x4 f32): lane m=lane&15, lanes<16 hold K={0,1},
// lanes>=16 hold K={2,3}. B mirrored with n=lane&15. C: elem r -> row
// r + ((lane>>4)<<3), col lane&15.
__global__ __launch_bounds__(256) void k_gemm_bias(
    const float* __restrict__ A, int lda,
    const float* __restrict__ B, int ldb, int K,
    const float* __restrict__ A2, int lda2,
    const float* __restrict__ B2, int K2,
    const float* __restrict__ bias,
    float* __restrict__ C, int ldc, int c_col_off, int Ntiles) {
    int wave = threadIdx.x >> 5;
    int lane = threadIdx.x & 31;
    int tile = blockIdx.x * 8 + wave;
    int mt = tile / Ntiles;
    int nt = tile - mt * Ntiles;
    int row0 = mt << 4;
    int col0 = nt << 4;
    int l15   = lane & 15;
    int khalf = (lane >> 4) << 1;   // 0 or 2
    int mhi   = (lane >> 4) << 3;   // 0 or 8

    float bv = bias[col0 + l15];
    v8f c;
#pragma unroll
    for (int r = 0; r < 8; ++r) c[r] = bv;

    {
        const float* ap = A + (size_t)(row0 + l15) * lda + khalf;
        const float* bp = B + (size_t)khalf * ldb + col0 + l15;
        for (int k = 0; k < K; k += 4) {
            v2f a, b;
            a.x = ap[0];  a.y = ap[1];
            b.x = bp[0];  b.y = bp[ldb];
            c = __builtin_amdgcn_wmma_f32_16x16x4_f32(
                    false, a, false, b, (short)0, c, false, false);
            ap += 4;
            bp += 4 * (size_t)ldb;
        }
    }
    if (A2) {  // uniform branch: EXEC stays all-1s around WMMA
        const float* ap = A2 + (size_t)(row0 + l15) * lda2 + khalf;
        const float* bp = B2 + (size_t)khalf * ldb + col0 + l15;
        for (int k = 0; k < K2; k += 4) {
            v2f a, b;
            a.x = ap[0];  a.y = ap[1];
            b.x = bp[0];  b.y = bp[ldb];
            c = __builtin_amdgcn_wmma_f32_16x16x4_f32(
                    false, a, false, b, (short)0, c, false, false);
            ap += 4;
            bp += 4 * (size_t)ldb;
        }
    }

    float* out = C + (size_t)(row0 + mhi) * ldc + c_col_off + col0 + l15;
#pragma unroll
    for (int r = 0; r < 8; ++r) out[(size_t)r * ldc] = c[r];
}

// ---------------------------------------------------------------------------
extern "C" void kernel_launch(void* const* d_in, const int* in_sizes, int n_in,
                              void* d_out, int out_size, void* d_ws, size_t ws_size,
                              hipStream_t stream) {
    const float* xCC    = (const float*)d_in[0];
    const float* xFace  = (const float*)d_in[1];
    const int*   src_cf = (const int*)d_in[2];
    const int*   dst_cf = (const int*)d_in[3];
    const float* ea_cf  = (const float*)d_in[4];
    const int*   src_fp = (const int*)d_in[5];
    const int*   dst_fp = (const int*)d_in[6];
    const float* ea_fp  = (const float*)d_in[7];
    const int*   src_pp = (const int*)d_in[8];
    const int*   dst_pp = (const int*)d_in[9];
    const float* ea_pp  = (const float*)d_in[10];
    const float* W_cf   = (const float*)d_in[11];
    const float* b_cf   = (const float*)d_in[12];
    const float* W_fp   = (const float*)d_in[13];
    const float* b_fp   = (const float*)d_in[14];
    const float* W_pp   = (const float*)d_in[15];
    const float* W_ppr  = (const float*)d_in[16];
    const float* b_pp   = (const float*)d_in[17];
    float* out = (float*)d_out;

    // workspace layout (floats)
    float* agg_cf = (float*)d_ws;                          // 100000*64
    float* xcat   = agg_cf + (size_t)N_FACES * C_IN;       // 100000*192
    float* agg_fp = xcat   + (size_t)N_FACES * C_CAT;      // 50000*192
    float* xp     = agg_fp + (size_t)N_POINTS * C_CAT;     // 50000*256
    float* agg_pp = xp     + (size_t)N_POINTS * C_OUT;     // 50000*256

    // zero the three scatter accumulators
    {
        int n4 = N_FACES * C_IN / 4;
        k_zero<<<(n4 + 255) / 256, 256, 0, stream>>>(agg_cf, n4);
        n4 = N_POINTS * C_CAT / 4;
        k_zero<<<(n4 + 255) / 256, 256, 0, stream>>>(agg_fp, n4);
        n4 = N_POINTS * C_OUT / 4;
        k_zero<<<(n4 + 255) / 256, 256, 0, stream>>>(agg_pp, n4);
    }

    // ---- stage 1: cell centers -> faces ----
    {
        int total = E_CF * (C_IN / 4);
        k_scatter<<<(total + 255) / 256, 256, 0, stream>>>(
            xCC, src_cf, dst_cf, ea_cf, agg_cf, E_CF, C_IN / 4, C_IN);
        int ntiles = C_CF / 16;                       // 8
        int blocks = (N_FACES / 16) * ntiles / 8;     // 6250
        k_gemm_bias<<<blocks, 256, 0, stream>>>(
            agg_cf, C_IN, W_cf, C_CF, C_IN,
            nullptr, 0, nullptr, 0,
            b_cf, xcat, C_CAT, 0, ntiles);
        int ct = N_FACES * 16;
        k_copy_face<<<(ct + 255) / 256, 256, 0, stream>>>(xFace, xcat);
    }

    // ---- stage 2: faces -> points ----
    {
        int total = E_FP * (C_CAT / 4);
        k_scatter<<<(total + 255) / 256, 256, 0, stream>>>(
            xcat, src_fp, dst_fp, ea_fp, agg_fp, E_FP, C_CAT / 4, C_CAT);
        int ntiles = C_OUT / 16;                      // 16
        int blocks = (N_POINTS / 16) * ntiles / 8;    // 6250
        k_gemm_bias<<<blocks, 256, 0, stream>>>(
            agg_fp, C_CAT, W_fp, C_OUT, C_CAT,
            nullptr, 0, nullptr, 0,
            b_fp, xp, C_OUT, 0, ntiles);
    }

    // ---- stage 3: points -> points, fused xp@W_pp_root + agg_pp@W_pp + b ----
    {
        int total = E_PP * (C_OUT / 4);
        k_scatter<<<(total + 255) / 256, 256, 0, stream>>>(
            xp, src_pp, dst_pp, ea_pp, agg_pp, E_PP, C_OUT / 4, C_OUT);
        int ntiles = C_OUT / 16;                      // 16
        int blocks = (N_POINTS / 16) * ntiles / 8;    // 6250
        k_gemm_bias<<<blocks, 256, 0, stream>>>(
            xp, C_OUT, W_ppr, C_OUT, C_OUT,
            agg_pp, C_OUT, W_pp, C_OUT,
            b_pp, out, C_OUT, 0, ntiles);
    }
}